// TensorProductOptimized_75548474737353
// MI455X (gfx1250) — compile-verified
//
#include <hip/hip_runtime.h>

typedef float v2f __attribute__((ext_vector_type(2)));
typedef float v8f __attribute__((ext_vector_type(8)));

// ---------------------------------------------------------------------------
// Compile-time real Clebsch-Gordan tables (verbatim port of the reference's
// Racah formula + e3nn real<->complex basis change, evaluated by constexpr).
// ---------------------------------------------------------------------------
namespace cgc {

struct CD { double re, im; };
constexpr CD cmul(CD a, CD b){ return CD{a.re*b.re - a.im*b.im, a.re*b.im + a.im*b.re}; }
constexpr CD cmulr(CD a, double b){ return CD{a.re*b, a.im*b}; }
constexpr CD cadd(CD a, CD b){ return CD{a.re+b.re, a.im+b.im}; }
constexpr CD cconj(CD a){ return CD{a.re, -a.im}; }

constexpr double fact(int n){ double r = 1.0; for (int i = 2; i <= n; ++i) r *= (double)i; return r; }
constexpr double csqrt(double x){
  if (x <= 0.0) return 0.0;
  double g = x > 1.0 ? x : 1.0;
  for (int i = 0; i < 80; ++i) g = 0.5*(g + x/g);
  return g;
}

constexpr double su2cg(int j1,int j2,int j3,int m1,int m2,int m3){
  if (m1 + m2 != m3) return 0.0;
  double pref = csqrt((2.0*j3+1.0)*fact(j3+j1-j2)*fact(j3-j1+j2)*fact(j1+j2-j3)/fact(j1+j2+j3+1));
  pref *= csqrt(fact(j3+m3)*fact(j3-m3)*fact(j1-m1)*fact(j1+m1)*fact(j2-m2)*fact(j2+m2));
  int kmin = 0;
  if (j2-j3-m1 > kmin) kmin = j2-j3-m1;
  if (j1-j3+m2 > kmin) kmin = j1-j3+m2;
  int kmax = j1+j2-j3;
  if (j1-m1 < kmax) kmax = j1-m1;
  if (j2+m2 < kmax) kmax = j2+m2;
  double s = 0.0;
  for (int k = kmin; k <= kmax; ++k){
    double d = fact(k)*fact(j1+j2-j3-k)*fact(j1-m1-k)*fact(j2+m2-k)*fact(j3-j2+m1+k)*fact(j3-j1-m2+k);
    s += ((k & 1) ? -1.0 : 1.0) / d;
  }
  return pref * s;
}

struct QM { CD q[9][9]; };
constexpr QM buildQ(int l){
  QM Q{};
  double is2 = csqrt(0.5);
  for (int m = -l; m < 0; ++m){
    Q.q[l+m][l-m] = CD{is2, 0.0};     // q[l+m, l+|m|] =  1/sqrt2
    Q.q[l+m][l+m] = CD{0.0, -is2};    // q[l+m, l-|m|] = -i/sqrt2
  }
  Q.q[l][l] = CD{1.0, 0.0};
  for (int m = 1; m <= l; ++m){
    double sg = (m & 1) ? -1.0 : 1.0;
    Q.q[l+m][l+m] = CD{sg*is2, 0.0};
    Q.q[l+m][l-m] = CD{0.0, sg*is2};
  }
  CD ph = (l%4==0) ? CD{1.0,0.0} : (l%4==1) ? CD{0.0,-1.0}
        : (l%4==2) ? CD{-1.0,0.0} : CD{0.0,1.0};        // (-i)^l
  for (int a = 0; a < 9; ++a)
    for (int b = 0; b < 9; ++b)
      Q.q[a][b] = cmul(Q.q[a][b], ph);
  return Q;
}

struct CGpad { float v[225]; };   // v[(i*5 + j)*9 + k]
constexpr CGpad buildCG(int l1,int l2,int l3){
  const int d1 = 2*l1+1, d2 = 2*l2+1, d3 = 2*l3+1;
  double C[5][5][9] = {};
  for (int i = 0; i < d1; ++i)
    for (int k = 0; k < d2; ++k)
      for (int n = 0; n < d3; ++n)
        C[i][k][n] = su2cg(l1, l2, l3, i-l1, k-l2, n-l3);
  QM Q1 = buildQ(l1); QM Q2 = buildQ(l2); QM Q3 = buildQ(l3);
  // staged einsum: result[a,b,c] = sum_{i,k,n} Q1[i,a] Q2[k,b] conj(Q3[n,c]) C[i,k,n]
  CD T1[5][5][9] = {};
  for (int a = 0; a < d1; ++a)
    for (int k = 0; k < d2; ++k)
      for (int n = 0; n < d3; ++n){
        CD acc{0.0,0.0};
        for (int i = 0; i < d1; ++i) acc = cadd(acc, cmulr(Q1.q[i][a], C[i][k][n]));
        T1[a][k][n] = acc;
      }
  CD T2[5][5][9] = {};
  for (int a = 0; a < d1; ++a)
    for (int b = 0; b < d2; ++b)
      for (int n = 0; n < d3; ++n){
        CD acc{0.0,0.0};
        for (int k = 0; k < d2; ++k) acc = cadd(acc, cmul(Q2.q[k][b], T1[a][k][n]));
        T2[a][b][n] = acc;
      }
  CGpad R{};
  for (int a = 0; a < d1; ++a)
    for (int b = 0; b < d2; ++b)
      for (int c = 0; c < d3; ++c){
        CD acc{0.0,0.0};
        for (int n = 0; n < d3; ++n) acc = cadd(acc, cmul(cconj(Q3.q[n][c]), T2[a][b][n]));
        R.v[(a*5 + b)*9 + c] = (float)acc.re;
      }
  return R;
}

} // namespace cgc

// 19 output chunks in e3nn-sorted order: key (l3, -p*(-1)^l3), stable by path
// enumeration. Each as its own __device__ const (separate constexpr evals).
__device__ const cgc::CGpad g_cg0  = cgc::buildCG(0,0,0);
__device__ const cgc::CGpad g_cg1  = cgc::buildCG(1,1,0);
__device__ const cgc::CGpad g_cg2  = cgc::buildCG(2,2,0);
__device__ const cgc::CGpad g_cg3  = cgc::buildCG(0,1,1);
__device__ const cgc::CGpad g_cg4  = cgc::buildCG(1,0,1);
__device__ const cgc::CGpad g_cg5  = cgc::buildCG(1,2,1);
__device__ const cgc::CGpad g_cg6  = cgc::buildCG(2,1,1);
__device__ const cgc::CGpad g_cg7  = cgc::buildCG(1,1,1);
__device__ const cgc::CGpad g_cg8  = cgc::buildCG(2,2,1);
__device__ const cgc::CGpad g_cg9  = cgc::buildCG(0,2,2);
__device__ const cgc::CGpad g_cg10 = cgc::buildCG(1,1,2);
__device__ const cgc::CGpad g_cg11 = cgc::buildCG(2,0,2);
__device__ const cgc::CGpad g_cg12 = cgc::buildCG(2,2,2);
__device__ const cgc::CGpad g_cg13 = cgc::buildCG(1,2,2);
__device__ const cgc::CGpad g_cg14 = cgc::buildCG(2,1,2);
__device__ const cgc::CGpad g_cg15 = cgc::buildCG(1,2,3);
__device__ const cgc::CGpad g_cg16 = cgc::buildCG(2,1,3);
__device__ const cgc::CGpad g_cg17 = cgc::buildCG(2,2,3);
__device__ const cgc::CGpad g_cg18 = cgc::buildCG(2,2,4);

__device__ inline const float* cgPtr(int c){
  switch (c){
    case 0:  return g_cg0.v;  case 1:  return g_cg1.v;  case 2:  return g_cg2.v;
    case 3:  return g_cg3.v;  case 4:  return g_cg4.v;  case 5:  return g_cg5.v;
    case 6:  return g_cg6.v;  case 7:  return g_cg7.v;  case 8:  return g_cg8.v;
    case 9:  return g_cg9.v;  case 10: return g_cg10.v; case 11: return g_cg11.v;
    case 12: return g_cg12.v; case 13: return g_cg13.v; case 14: return g_cg14.v;
    case 15: return g_cg15.v; case 16: return g_cg16.v; case 17: return g_cg17.v;
    default: return g_cg18.v;
  }
}

struct Meta {
  int l1[19], l2[19], l3[19], obase[19], abase[19];
  int chunkOfO[81];
  int AS[96];              // (element-column offset within row)<<4 | (2*l3+1)
};
constexpr Meta buildMeta(){
  Meta M{};
  const int L1[19] = {0,1,2, 0,1,1,2, 1,2, 0,1,2,2, 1,2, 1,2, 2, 2};
  const int L2[19] = {0,1,2, 1,0,2,1, 1,2, 2,1,0,2, 2,1, 2,1, 2, 2};
  const int L3[19] = {0,0,0, 1,1,1,1, 1,1, 2,2,2,2, 2,2, 3,3, 3, 4};
  int ob = 0, ab = 0;
  for (int c = 0; c < 19; ++c){
    M.l1[c] = L1[c]; M.l2[c] = L2[c]; M.l3[c] = L3[c];
    M.obase[c] = ob; M.abase[c] = ab;
    const int S = 2*L3[c] + 1;
    for (int k = 0; k < S; ++k){
      M.chunkOfO[ob + k] = c;
      M.AS[ob + k] = ((ab + k) << 4) | S;
    }
    ob += S; ab += 128*S;   // ends at ob=81, ab=10368
  }
  for (int o = 81; o < 96; ++o) M.AS[o] = 0;
  return M;
}
__device__ const Meta g_meta = buildMeta();

// ---------------------------------------------------------------------------
// Main kernel: one workgroup (128 threads, 4 wave32) per sample n.
//   out[128x81] = X1[128x9(+pad 12)] @ W[9(+pad)x81(+pad 96)]  via
//   v_wmma_f32_16x16x4_f32, 8 M-tiles x 6 N-tiles x 3 K-steps.
// Memory-bound op (~680 MB NT stores + ~76 MB loads -> ~32 us at 23.3 TB/s);
// WMMA keeps the compute stream short; stores use SGPR-base + 32-bit VGPR
// offset (GVS) addressing to minimize per-store VALU work.
// ---------------------------------------------------------------------------
__global__ __launch_bounds__(128) void tp_wmma_f32_kernel(
    const float* __restrict__ x1_0, const float* __restrict__ x1_1, const float* __restrict__ x1_2,
    const float* __restrict__ x2_0, const float* __restrict__ x2_1, const float* __restrict__ x2_2,
    float* __restrict__ out)
{
  __shared__ __align__(16) float Xs[128*12];   // X[u][k], pitch 12 (k=9..11 zero)
  __shared__ __align__(16) float Ws[96*14];    // W[o][k], pitch 14 (k=9..11 zero)

  const int n = blockIdx.x;
  const int t = threadIdx.x;

  // ---- stage x1 into LDS, fully coalesced (read-once data: NT loads) ----
  Xs[t*12 + 0]  = __builtin_nontemporal_load(&x1_0[(unsigned)(n*128 + t)]);
  Xs[t*12 + 9]  = 0.0f;
  Xs[t*12 + 10] = 0.0f;
  Xs[t*12 + 11] = 0.0f;
  {
    const unsigned b1 = (unsigned)n * 384u;
    #pragma unroll
    for (int r = 0; r < 3; ++r){
      const int tt = t + 128*r;
      const float v = __builtin_nontemporal_load(&x1_1[b1 + (unsigned)tt]);
      Xs[(tt/3)*12 + 1 + (tt%3)] = v;
    }
    const unsigned b2 = (unsigned)n * 640u;
    #pragma unroll
    for (int r = 0; r < 5; ++r){
      const int tt = t + 128*r;
      const float v = __builtin_nontemporal_load(&x1_2[b2 + (unsigned)tt]);
      Xs[(tt/5)*12 + 4 + (tt%5)] = v;
    }
  }

  // ---- build per-sample weight matrix W[i,o] = sum_j CG[i,j,o] * x2[j] ----
  if (t < 96){
    float w[12] = {0,0,0,0,0,0,0,0,0,0,0,0};
    if (t < 81){
      const int c    = g_meta.chunkOfO[t];
      const int k3   = t - g_meta.obase[c];
      const int l1   = g_meta.l1[c];
      const int l2   = g_meta.l2[c];
      const int ioff = (l1 == 0) ? 0 : (l1 == 1) ? 1 : 4;
      const int d1 = 2*l1 + 1, d2 = 2*l2 + 1;
      float x2v[5];
      if (l2 == 0)      { x2v[0] = x2_0[n]; }
      else if (l2 == 1) { for (int j = 0; j < 3; ++j) x2v[j] = x2_1[(unsigned)(n*3 + j)]; }
      else              { for (int j = 0; j < 5; ++j) x2v[j] = x2_2[(unsigned)(n*5 + j)]; }
      const float* cgv = cgPtr(c);
      for (int i = 0; i < d1; ++i){
        float s = 0.0f;
        for (int j = 0; j < d2; ++j) s += x2v[j] * cgv[(i*5 + j)*9 + k3];
        w[ioff + i] = s;
      }
    }
    #pragma unroll
    for (int k = 0; k < 12; k += 2){
      v2f p; p.x = w[k]; p.y = w[k+1];
      *(v2f*)&Ws[t*14 + k] = p;
    }
  }
  __syncthreads();

  const int lane = t & 31;
  const int wid  = t >> 5;
  const int lo   = lane & 15;
  const int hi   = lane >> 4;

  // A operand regs (layout: lane = M + 16*(K>>1), vgpr = K&1): per wave 2 M-tiles.
  v2f A[2][3];
  #pragma unroll
  for (int mt = 0; mt < 2; ++mt){
    const int m = wid*32 + mt*16 + lo;
    #pragma unroll
    for (int s = 0; s < 3; ++s){
      const int k = s*4 + 2*hi;
      A[mt][s] = *(const v2f*)&Xs[m*12 + k];
    }
  }

  float* __restrict__ outn = out + (size_t)n * 10368;   // wave-uniform SGPR base

  #pragma unroll
  for (int nt = 0; nt < 6; ++nt){
    const int o  = nt*16 + lo;
    const int as = g_meta.AS[o];
    const int Ao = as >> 4;
    const int So = as & 15;
    // B operand (B[k,n]: lane = n + 16*(k>>1), vgpr = k&1) -> b64 from W[o][k..k+1]
    v2f B[3];
    #pragma unroll
    for (int s = 0; s < 3; ++s){
      const int k = s*4 + 2*hi;
      B[s] = *(const v2f*)&Ws[o*14 + k];
    }
    #pragma unroll
    for (int mt = 0; mt < 2; ++mt){
      v8f acc = {0.f,0.f,0.f,0.f,0.f,0.f,0.f,0.f};
      acc = __builtin_amdgcn_wmma_f32_16x16x4_f32(false, A[mt][0], false, B[0], (short)0, acc, false, false);
      acc = __builtin_amdgcn_wmma_f32_16x16x4_f32(false, A[mt][1], false, B[1], (short)0, acc, false, false);
      acc = __builtin_amdgcn_wmma_f32_16x16x4_f32(false, A[mt][2], false, B[2], (short)0, acc, false, false);
      if (o < 81){
        // C/D layout: vgpr r -> M = m0 + r + 8*hi, N = o. Streaming NT stores,
        // 32-bit strength-reduced offsets against the uniform row base.
        const int ub = wid*32 + mt*16 + 8*hi;
        unsigned idx = (unsigned)(Ao + ub * So);
        #pragma unroll
        for (int r = 0; r < 8; ++r){
          __builtin_nontemporal_store(acc[r], outn + idx);
          idx += (unsigned)So;
        }
      }
    }
  }
}

extern "C" void kernel_launch(void* const* d_in, const int* in_sizes, int n_in,
                              void* d_out, int out_size, void* d_ws, size_t ws_size,
                              hipStream_t stream) {
  (void)n_in; (void)out_size; (void)d_ws; (void)ws_size;
  const float* x1_0 = (const float*)d_in[0];
  const float* x1_1 = (const float*)d_in[1];
  const float* x1_2 = (const float*)d_in[2];
  const float* x2_0 = (const float*)d_in[3];
  const float* x2_1 = (const float*)d_in[4];
  const float* x2_2 = (const float*)d_in[5];
  float* out = (float*)d_out;
  const int N = in_sizes[0] / 128;   // x1_0 is [N,128,1]
  tp_wmma_f32_kernel<<<dim3(N), dim3(128), 0, stream>>>(x1_0, x1_1, x1_2, x2_0, x2_1, x2_2, out);
}